// DTWLayer_14259291422990
// MI455X (gfx1250) — compile-verified
//
#include <hip/hip_runtime.h>

// DTW layer for MI455X (gfx1250, wave32).
//
// Strategy: one wave (32 lanes) per (b, n) pair.
//  - cross[p, j] = sum_d patts[n,d,p] * x[b,d,j] computed with
//    V_WMMA_F32_16X16X4_F32 (K=12 = 3 chained K=4 steps, M=32 = 2 tiles),
//    streamed over 16-column tiles of j. cost = p2 + x2 - 2*cross.
//  - DP column cur[i] = c[i] + min(a[i], w*cur[i-1]) with w = RHO^(1/32) = 1:
//    each cell is f_i(v) = min(alpha_i, beta_i + v); these compose closedly,
//    so a 5-step Hillis-Steele scan (ds_bpermute) solves a whole column.
//    (For RHO != 1 a per-segment weight would also need to be tracked.)
//  - Fusion keeps the 128 MB cost tensor out of HBM entirely; traffic is
//    ~1.6 MB in + 4 MB out -> latency bound, which is optimal here.

typedef float v2f __attribute__((ext_vector_type(2)));
typedef float v8f __attribute__((ext_vector_type(8)));

#define B_    32
#define N_    32
#define D_    12
#define P_    32
#define L_    1024
#define LOUT_ 32
#define BIGV  1e30f

__device__ __forceinline__ float bperm_f(int src_lane, float v) {
    int r = __builtin_amdgcn_ds_bpermute(src_lane << 2,
                                         __builtin_bit_cast(int, v));
    return __builtin_bit_cast(float, r);
}

__global__ __launch_bounds__(32) void dtw_wmma_kernel(
    const float* __restrict__ x,      // [B, D, L]
    const float* __restrict__ patts,  // [N, D, P]
    float* __restrict__ out)          // [B, N, P, LOUT]
{
    const int lane = threadIdx.x;   // 0..31 ; lane == DP row p
    const int hi   = lane >> 4;     // half-wave select (WMMA layouts)
    const int lo   = lane & 15;
    const int b    = blockIdx.x;
    const int n    = blockIdx.y;

    __shared__ float p2s[P_];
    __shared__ float ctile[P_ * 17];  // 32 rows x 16 cols, stride 17 (no bank conflicts)

    const float* Pn = patts + n * (D_ * P_);
    const float* Xb = x + b * (D_ * L_);

    // p2[p] = sum_d patts[n,d,p]^2, one p per lane.
    {
        float s = 0.0f;
        #pragma unroll
        for (int d = 0; d < D_; ++d) { float v = Pn[d * P_ + lane]; s += v * v; }
        p2s[lane] = s;
    }
    __syncthreads();

    // A fragments (loop invariant): A[M=p, K=d] = patts[n, d, p].
    // f32 16x16x4 A layout: lane -> M = lane%16 ; vgpr r -> K = 2*hi + r.
    v2f Afrag[2][3];
    #pragma unroll
    for (int m = 0; m < 2; ++m)
        #pragma unroll
        for (int kc = 0; kc < 3; ++kc) {
            int d0 = 4 * kc + 2 * hi;
            int p  = m * 16 + lo;
            v2f a; a.x = Pn[d0 * P_ + p]; a.y = Pn[(d0 + 1) * P_ + p];
            Afrag[m][kc] = a;
        }

    // p2 values matching this lane's C-layout rows: M = m*16 + 8*hi + r.
    float p2v[2][8];
    #pragma unroll
    for (int m = 0; m < 2; ++m)
        #pragma unroll
        for (int r = 0; r < 8; ++r)
            p2v[m][r] = p2s[m * 16 + 8 * hi + r];

    float dprev = BIGV;  // D[lane, j-1]
    float* outp = out + ((b * N_ + n) * P_ + lane) * LOUT_;
    const float INF = __builtin_inff();

    for (int jt = 0; jt < L_ / 16; ++jt) {
        const int jbase = jt * 16;
        const int j     = jbase + lo;

        // B fragments: B[K=d, N=j] = x[b, d, j]; same half-wave K mapping as A.
        v2f Bfrag[3];
        float sx = 0.0f;  // partial x2[j] over this half-wave's 6 d-values
        #pragma unroll
        for (int kc = 0; kc < 3; ++kc) {
            int d0 = 4 * kc + 2 * hi;
            float b0 = Xb[d0 * L_ + j];
            float b1 = Xb[(d0 + 1) * L_ + j];
            v2f bb; bb.x = b0; bb.y = b1;
            Bfrag[kc] = bb;
            sx += b0 * b0 + b1 * b1;
        }
        // full x2[j]: add partner half-wave's 6 d-values (lane ^ 16).
        sx += bperm_f(lane ^ 16, sx);

        if (jt + 1 < L_ / 16)  // gfx1250 global_prefetch_b8 on the x stream
            __builtin_prefetch(&Xb[(2 * hi) * L_ + j + 16], 0, 3);

        // cross tiles via WMMA, convert to cost, transpose through LDS.
        #pragma unroll
        for (int m = 0; m < 2; ++m) {
            v8f acc = {};
            acc = __builtin_amdgcn_wmma_f32_16x16x4_f32(false, Afrag[m][0],
                    false, Bfrag[0], (short)0, acc, false, false);
            acc = __builtin_amdgcn_wmma_f32_16x16x4_f32(false, Afrag[m][1],
                    false, Bfrag[1], (short)0, acc, false, false);
            acc = __builtin_amdgcn_wmma_f32_16x16x4_f32(false, Afrag[m][2],
                    false, Bfrag[2], (short)0, acc, false, false);
            #pragma unroll
            for (int r = 0; r < 8; ++r) {
                float cst = p2v[m][r] + sx - 2.0f * acc[r];
                ctile[(m * 16 + 8 * hi + r) * 17 + lo] = cst;  // row-major in LDS
            }
        }
        __syncthreads();

        // DP over the 16 columns of this tile.
        for (int jj = 0; jj < 16; ++jj) {
            const int   jg = jbase + jj;            // uniform
            const float c  = ctile[lane * 17 + jj]; // cost[lane, jg]

            float a;
            if (jg == 0) {
                a = (lane == 0) ? 0.0f : BIGV;      // first column boundary
            } else {
                float up = bperm_f(lane - 1, dprev); // D[lane-1, j-1]
                up = (lane == 0) ? BIGV : up;
                a  = fminf(up, dprev);               // min(D[i-1,j-1], w*D[i,j-1]), w=1
            }

            // inclusive affine-min scan: F_i = f_i o ... o f_0, f_i(v)=min(al,be+v)
            float al = c + a;
            float be = c;
            #pragma unroll
            for (int o = 1; o < 32; o <<= 1) {
                float af = bperm_f(lane - o, al);
                float bf = bperm_f(lane - o, be);
                bool ok = (lane >= o);
                af = ok ? af : INF;    // identity function below segment start
                bf = ok ? bf : 0.0f;
                al = fminf(al, be + af);
                be = be + bf;
            }
            dprev = fminf(al, be + BIGV);  // cur[i] = F_i(BIG)

            if (jg >= L_ - LOUT_) outp[jg - (L_ - LOUT_)] = dprev;
        }
        __syncthreads();
    }
}

extern "C" void kernel_launch(void* const* d_in, const int* in_sizes, int n_in,
                              void* d_out, int out_size, void* d_ws, size_t ws_size,
                              hipStream_t stream) {
    const float* x     = (const float*)d_in[0];  // [32, 12, 1024]
    const float* patts = (const float*)d_in[1];  // [32, 12, 32]
    float*       outp  = (float*)d_out;          // [32, 32, 32, 32]
    (void)in_sizes; (void)n_in; (void)out_size; (void)d_ws; (void)ws_size;

    dim3 grid(B_, N_);
    dim3 block(32);
    dtw_wmma_kernel<<<grid, block, 0, stream>>>(x, patts, outp);
}